// GraphConv_86277303042053
// MI455X (gfx1250) — compile-verified
//
#include <hip/hip_runtime.h>
#include <hip/hip_bf16.h>

// GraphConv on MI455X (gfx1250, wave32):
//   agg = scatter_add(nodes[senders] * edges, receivers)      [N,128]  (L2-resident atomics)
//   out = agg @ W + nodes @ W_root + b                        [N,128]  (V_WMMA_F32_16X16X4_F32)

#define N_NODES 100000
#define FDIM    128
#define ODIM    128

typedef __attribute__((ext_vector_type(2))) float v2f;
typedef __attribute__((ext_vector_type(8))) float v8f;

// ---------------------------------------------------------------------------
// Kernel 1: zero the aggregation workspace (harness poisons d_ws with 0xAA).
// ---------------------------------------------------------------------------
__global__ __launch_bounds__(256) void gc_zero_kernel(float4* __restrict__ agg4, int n4) {
    int i = blockIdx.x * blockDim.x + threadIdx.x;
    if (i < n4) agg4[i] = make_float4(0.f, 0.f, 0.f, 0.f);
}

// ---------------------------------------------------------------------------
// Kernel 2: gather + scatter-add. One wave32 per edge; each lane handles 4
// consecutive features (float4 gather, 4x global_atomic_add_f32 scatter).
// nodes (51MB) + agg (51MB) both fit in the 192MB L2, so the atomics resolve
// in-L2 rather than at HBM.
// ---------------------------------------------------------------------------
__global__ __launch_bounds__(256) void gc_scatter_kernel(
    const float* __restrict__ nodes,
    const int*   __restrict__ senders,
    const int*   __restrict__ receivers,
    const float* __restrict__ edges,
    float*       __restrict__ agg,
    int E)
{
    int wave = (blockIdx.x * blockDim.x + threadIdx.x) >> 5;
    int lane = threadIdx.x & 31;
    if (wave >= E) return;

    int s = senders[wave];
    int r = receivers[wave];
    float w = edges[wave];

    const float4* src = (const float4*)(nodes + (size_t)s * FDIM);
    float4 v = src[lane];                       // 4 consecutive floats per lane
    float* dst = agg + (size_t)r * FDIM + lane * 4;
    atomicAdd(dst + 0, v.x * w);
    atomicAdd(dst + 1, v.y * w);
    atomicAdd(dst + 2, v.z * w);
    atomicAdd(dst + 3, v.w * w);
}

// ---------------------------------------------------------------------------
// Kernel 3: fused dual GEMM + bias via V_WMMA_F32_16X16X4_F32.
// One wave per 16x16 output tile; K=128 in steps of 4, accumulating both
// agg@W and nodes@W_root into a single v8f accumulator (linearity).
//
// Layouts per CDNA5 ISA 7.12.2:
//   A 16x4 : lanes 0-15 -> M=lane, VGPR0/1 = K=0/1; lanes 16-31 -> K=2/3
//   B 4x16 : lanes 0-15 -> rows K=0/1, lanes 16-31 -> rows K=2/3, N=lane%16
//   C/D    : VGPR r -> row r (lanes 0-15) / row r+8 (lanes 16-31), N=lane%16
// ---------------------------------------------------------------------------
__global__ __launch_bounds__(128) void gc_gemm_kernel(
    const float* __restrict__ agg,
    const float* __restrict__ nodes,
    const float* __restrict__ W,
    const float* __restrict__ Wroot,
    const float* __restrict__ bias,
    float*       __restrict__ out,
    int numTiles)
{
    const int wavesPerBlock = blockDim.x >> 5;
    int tile = blockIdx.x * wavesPerBlock + (threadIdx.x >> 5);
    if (tile >= numTiles) return;

    int lane    = threadIdx.x & 31;
    int halfSel = lane >> 4;       // 0: lanes 0-15, 1: lanes 16-31
    int laneM   = lane & 15;       // A-row offset / B-col offset

    int rowTile = tile >> 3;       // ODIM/16 = 8 column tiles
    int colTile = tile & 7;
    int row0 = rowTile * 16;
    int col0 = colTile * 16;

    const float* Arow = agg   + (size_t)(row0 + laneM) * FDIM;
    const float* Nrow = nodes + (size_t)(row0 + laneM) * FDIM;

    v8f acc = {0.f, 0.f, 0.f, 0.f, 0.f, 0.f, 0.f, 0.f};

    // ---- agg @ W ----
    #pragma unroll 4
    for (int k = 0; k < FDIM; k += 4) {
        int ka = k + halfSel * 2;
        v2f a; a.x = Arow[ka]; a.y = Arow[ka + 1];
        v2f b; b.x = W[(size_t)ka * ODIM + col0 + laneM];
               b.y = W[(size_t)(ka + 1) * ODIM + col0 + laneM];
        acc = __builtin_amdgcn_wmma_f32_16x16x4_f32(
            false, a, false, b, (short)0, acc, false, false);
    }
    // ---- nodes @ W_root ----
    #pragma unroll 4
    for (int k = 0; k < FDIM; k += 4) {
        int ka = k + halfSel * 2;
        v2f a; a.x = Nrow[ka]; a.y = Nrow[ka + 1];
        v2f b; b.x = Wroot[(size_t)ka * ODIM + col0 + laneM];
               b.y = Wroot[(size_t)(ka + 1) * ODIM + col0 + laneM];
        acc = __builtin_amdgcn_wmma_f32_16x16x4_f32(
            false, a, false, b, (short)0, acc, false, false);
    }

    float bv = bias[col0 + laneM];

    #pragma unroll
    for (int r = 0; r < 8; ++r) {
        int row = row0 + r + halfSel * 8;
        out[(size_t)row * ODIM + col0 + laneM] = acc[r] + bv;
    }
}

// ---------------------------------------------------------------------------
// Launch. Input order: nodes, senders, receivers, edges, W, b, W_root.
// d_ws must hold agg: N*128*4 = 51.2 MB.
// ---------------------------------------------------------------------------
extern "C" void kernel_launch(void* const* d_in, const int* in_sizes, int n_in,
                              void* d_out, int out_size, void* d_ws, size_t ws_size,
                              hipStream_t stream) {
    const float* nodes     = (const float*)d_in[0];
    const int*   senders   = (const int*)  d_in[1];
    const int*   receivers = (const int*)  d_in[2];
    const float* edges     = (const float*)d_in[3];
    const float* W         = (const float*)d_in[4];
    const float* bias      = (const float*)d_in[5];
    const float* Wroot     = (const float*)d_in[6];
    float*       out       = (float*)d_out;
    float*       agg       = (float*)d_ws;          // [N, 128] scratch

    const int E = in_sizes[1];                      // 1.6M edges
    const int nAgg  = N_NODES * FDIM;               // 12.8M floats
    const int nAgg4 = nAgg / 4;

    // 1) zero agg
    gc_zero_kernel<<<(nAgg4 + 255) / 256, 256, 0, stream>>>((float4*)agg, nAgg4);

    // 2) gather * edge weight, scatter-add (one wave per edge -> 8 edges/block)
    int edgeBlocks = (E + 7) / 8;
    gc_scatter_kernel<<<edgeBlocks, 256, 0, stream>>>(
        nodes, senders, receivers, edges, agg, E);

    // 3) fused WMMA GEMM: out = agg@W + nodes@W_root + b
    int numTiles = (N_NODES / 16) * (ODIM / 16);    // 6250 * 8 = 50000
    int gemmBlocks = (numTiles + 3) / 4;            // 4 waves / 128-thread block
    gc_gemm_kernel<<<gemmBlocks, 128, 0, stream>>>(
        agg, nodes, W, Wroot, bias, out, numTiles);
}